// ContiguousMultichannelSampling_61495341744240
// MI455X (gfx1250) — compile-verified
//
#include <hip/hip_runtime.h>
#include <hip/hip_bf16.h>
#include <stdint.h>

// ---------------------------------------------------------------------------
// Top-p (nucleus) sampling, CDNA5 / gfx1250.
//
// 512 rows x 50257 vocab. HBM floor = 103 MB / 23.3 TB/s ~= 4.4 us.
// One workgroup per row: stream the whole 201 KB row into the 320 KB LDS
// ONCE (TDM tensor_load_to_lds if available, else async global->LDS b128),
// then do all top-p refinement (2-level 2048-bucket float-bit histograms +
// block scans + ordinal selection) entirely out of LDS. No sort, one HBM pass.
// ---------------------------------------------------------------------------

#define NT      512                 // threads per block (16 waves of 32)
#define NB      2048                // histogram buckets per level
#define VOCAB   50257
#define ROWPAD  50260               // VOCAB + 3 slack for alignment shift
#define NCH     2
#define BSZ     32
#define BEAM    8
#define TOPP_C  0.95f

// LDS: row(201040) + h1(16K) + h2(16K) + scan bufs(4K) + bcast(64) = 237,968 B
#define SMEM_BYTES ((ROWPAD + 4*NB + 2*NT + 16) * 4)

typedef int v4i __attribute__((ext_vector_type(4)));
typedef __attribute__((address_space(1))) v4i gv4i;
typedef __attribute__((address_space(3))) v4i lv4i;

typedef unsigned int u32x4 __attribute__((ext_vector_type(4)));
typedef int          i32x4 __attribute__((ext_vector_type(4)));
typedef int          i32x8 __attribute__((ext_vector_type(8)));

#if __has_builtin(__builtin_amdgcn_tensor_load_to_lds)
  #define HAVE_TDM 1
  #if __has_include(<hip/amd_detail/amd_gfx1250_TDM.h>)
    #define TDM_6ARG 1            // therock-10.0 headers => 6-arg builtin
  #endif
#endif

// ---- fallback: async global->LDS copy of one 16B chunk --------------------
__device__ __forceinline__ void async_copy16(const void* gsrc, void* ldst) {
#if __has_builtin(__builtin_amdgcn_global_load_async_to_lds_b128)
  __builtin_amdgcn_global_load_async_to_lds_b128(
      (gv4i*)(unsigned long long)(uintptr_t)gsrc,
      (lv4i*)(unsigned)(uintptr_t)ldst,
      0, 0);
#else
  *(float4*)ldst = *(const float4*)gsrc;
#endif
}

__device__ __forceinline__ void wait_async0() {
#if __has_builtin(__builtin_amdgcn_s_wait_asynccnt)
  __builtin_amdgcn_s_wait_asynccnt(0);
#else
  asm volatile("s_wait_asynccnt 0" ::: "memory");
#endif
}

__device__ __forceinline__ void wait_tensor0() {
#if __has_builtin(__builtin_amdgcn_s_wait_tensorcnt)
  __builtin_amdgcn_s_wait_tensorcnt(0);
#else
  asm volatile("s_wait_tensorcnt 0" ::: "memory");
#endif
}

#ifdef HAVE_TDM
// ---- one-shot TDM DMA: 1 x VOCAB tile of 4-byte elements, global -> LDS ---
// D# bitfields per CDNA5 ISA ch.8 (group0: count/lds_addr/global_addr/type,
// group1: data_size, tensor_dim0/1, tile_dim0/1/2, strides).
__device__ __forceinline__ void tdm_load_row(const float* gsrc, void* ldst) {
  unsigned long long ga = (unsigned long long)(uintptr_t)gsrc;
  unsigned lds = (unsigned)(uintptr_t)ldst;
  u32x4 g0;
  g0.x = 1u;                                          // count=1 (valid user D#)
  g0.y = lds;                                         // lds_addr (bytes)
  g0.z = (unsigned)(ga & 0xFFFFFFFFu);                // global_addr[31:0]
  g0.w = (unsigned)((ga >> 32) & 0x1FFFFFFu)          // global_addr[56:32]
       | (2u << 30);                                  // type=2 ("image")
  i32x8 g1;
  g1.s0 = (int)(2u << 16);                            // data_size=2 -> 4B elems
  g1.s1 = (int)((VOCAB & 0xFFFFu) << 16);             // tensor_dim0[15:0]
  g1.s2 = (int)(((VOCAB >> 16) & 0xFFFFu)             // tensor_dim0[31:16]
       | (1u << 16));                                 // tensor_dim1 = 1
  g1.s3 = (int)((VOCAB & 0xFFFFu) << 16);             // tile_dim0 = VOCAB
  g1.s4 = 1;                                          // tile_dim1=1, tile_dim2=0
  g1.s5 = VOCAB;                                      // tensor_dim0_stride[31:0]
  g1.s6 = 0;
  g1.s7 = 0;
  i32x4 g2 = {0, 0, 0, 0};                            // dims 2/3 unused
  i32x4 g3 = {0, 0, 0, 0};
#ifdef TDM_6ARG
  i32x8 g4 = {0, 0, 0, 0, 0, 0, 0, 0};
  __builtin_amdgcn_tensor_load_to_lds(g0, g1, g2, g3, g4, 0);
#else
  __builtin_amdgcn_tensor_load_to_lds(g0, g1, g2, g3, 0);
#endif
}
#endif

// ---- block-wide inclusive Hillis-Steele scans over NT entries -------------
__device__ __forceinline__ void scan_incl_f(float* buf, int tid) {
  for (int off = 1; off < NT; off <<= 1) {
    float x = (tid >= off) ? buf[tid - off] : 0.0f;
    __syncthreads();
    buf[tid] += x;
    __syncthreads();
  }
}
__device__ __forceinline__ void scan_incl_i(int* buf, int tid) {
  for (int off = 1; off < NT; off <<= 1) {
    int x = (tid >= off) ? buf[tid - off] : 0;
    __syncthreads();
    buf[tid] += x;
    __syncthreads();
  }
}

// ---- descend a 2048-bucket histogram from the top until cum >= target -----
__device__ __forceinline__ void descend2048(const float* hs, float target,
                                            float* spart, float* bcf, int* bci,
                                            int tid, int* outB, float* outAb) {
  __syncthreads();
  if (tid == 0) { bci[0] = NB - 1; bcf[0] = 0.0f; }   // fallback
  const int base = (NB - 1) - 4 * tid;                // 4 buckets/thread, desc
  float v0 = hs[base], v1 = hs[base-1], v2 = hs[base-2], v3 = hs[base-3];
  float local = (v0 + v1) + (v2 + v3);
  spart[tid] = local;
  __syncthreads();
  scan_incl_f(spart, tid);
  float incl = spart[tid];
  float excl = incl - local;
  if (excl < target && incl >= target) {              // crossing in my chunk
    float acc = excl; int bsel;
    if (acc + v0 >= target) bsel = base;
    else { acc += v0;
      if (acc + v1 >= target) bsel = base - 1;
      else { acc += v1;
        if (acc + v2 >= target) bsel = base - 2;
        else { acc += v2; bsel = base - 3; } } }
    bci[0] = bsel; bcf[0] = acc;
  }
  __syncthreads();
  *outB = bci[0]; *outAb = bcf[0];
  __syncthreads();
}

// ---- second-level histogram (float bits [18:8]) of one level-1 bucket -----
__device__ __forceinline__ void build_h2(const float* row, float* h2s, int* h2c,
                                         unsigned Bhi, int tid) {
  __syncthreads();
  for (int i = tid; i < NB; i += NT) { h2s[i] = 0.0f; h2c[i] = 0; }
  __syncthreads();
  for (int i = tid; i < VOCAB; i += NT) {
    float p = row[i];
    unsigned kb = __float_as_uint(p);
    if ((kb >> 19) == Bhi) {
      int k2 = (int)((kb >> 8) & (NB - 1));
      atomicAdd(&h2c[k2], 1);        // ds_add_u32
      atomicAdd(&h2s[k2], p);        // ds_add_f32
    }
  }
  __syncthreads();
}

__global__ __launch_bounds__(NT)
void topp_sample_kernel(const float* __restrict__ lprobs,
                        const float* __restrict__ scores,
                        const float* __restrict__ uvec,
                        const int*   __restrict__ step_p,
                        float* __restrict__ out,
                        int sdim) {
  extern __shared__ unsigned char smem[];
  float* storage = (float*)smem;                 // ROWPAD floats
  float* h1s  = storage + ROWPAD;                // NB floats
  int*   h1c  = (int*)(h1s + NB);                // NB ints
  float* h2s  = (float*)(h1c + NB);              // NB floats
  int*   h2c  = (int*)(h2s + NB);                // NB ints
  float* spart  = (float*)(h2c + NB);            // NT floats
  int*   sparti = (int*)(spart + NT);            // NT ints
  float* bcf  = (float*)(sparti + NT);           // 8 floats broadcast
  int*   bci  = (int*)(bcf + 8);                 // 8 ints broadcast

  const int r   = blockIdx.x;                    // row = ((ch*32+b)*8+bm)
  const int ch  = r >> 8;
  const int b   = (r >> 3) & 31;
  const int bm  = r & 7;
  const int tid = threadIdx.x;

  const float* g = lprobs + (size_t)r * VOCAB;

  // ---- Phase A: single bulk copy of the row into LDS ----------------------
#ifdef HAVE_TDM
  float* row = storage;                          // TDM handles 4B alignment
  for (int i = tid; i < NB; i += NT) { h1s[i] = 0.0f; h1c[i] = 0; } // overlap
  if (tid < 32) {                                // wave 0 issues one DMA
    tdm_load_row(g, storage);
    wait_tensor0();
  }
  __syncthreads();
#else
  // Row r starts at byte r*201028 (== 4*r mod 16). Shift the LDS image by
  // (r mod 4) elements so every B128 chunk is 16B-aligned in BOTH spaces.
  const int shift  = r & 3;
  float* row = storage + shift;                  // row[i] <-> g[i]
  const int kstart = (4 - shift) & 3;            // first 16B-aligned element
  const int nfull  = (VOCAB - kstart) >> 2;      // # of 16B chunks
  for (int c = tid; c < nfull; c += NT) {
    const int k = kstart + 4 * c;
    async_copy16(g + k, row + k);
  }
  if (tid == 0) {                                // <=6 head/tail elements
    for (int k = 0; k < kstart; ++k) row[k] = g[k];
    for (int k = kstart + 4 * nfull; k < VOCAB; ++k) row[k] = g[k];
  }
  for (int i = tid; i < NB; i += NT) { h1s[i] = 0.0f; h1c[i] = 0; } // overlap
  wait_async0();
  __syncthreads();
#endif

  // ---- Phase B: exp + level-1 histogram on float bits [30:19] -------------
  for (int i = tid; i < VOCAB; i += NT) {
    float p = __expf(row[i]);                    // p in (0,1]
    row[i] = p;                                  // keep probs in LDS
    int k = (int)(__float_as_uint(p) >> 19);     // 0..2032 < NB
    atomicAdd(&h1c[k], 1);
    atomicAdd(&h1s[k], p);
  }
  __syncthreads();

  // ---- Phase C/D/E: trim boundary (two-level descent) ---------------------
  int B1; float above1;
  descend2048(h1s, TOPP_C, spart, bcf, bci, tid, &B1, &above1);

  build_h2(row, h2s, h2c, (unsigned)B1, tid);
  float need1 = TOPP_C - above1;
  int B2; float above2;
  descend2048(h2s, fmaxf(need1, 1e-30f), spart, bcf, bci, tid, &B2, &above2);

  int   n3 = h2c[B2]; if (n3 < 1) n3 = 1;
  float s3 = h2s[B2];
  float vavg = s3 / (float)n3; if (!(vavg > 0.0f)) vavg = 1e-30f;
  float need2 = need1 - above2;
  int m = (int)floorf(need2 / vavg) + 1;         // first element reaching topp
  if (m < 1) m = 1; if (m > n3) m = n3;
  float S = above1 + above2 + (float)m * vavg;   // trimmed total

  // ---- Phase F: weighted selection at threshold t = u * S -----------------
  float t = fmaxf(uvec[r] * S, 1e-30f);
  int Bs; float aboveS;
  descend2048(h1s, t, spart, bcf, bci, tid, &Bs, &aboveS);
  build_h2(row, h2s, h2c, (unsigned)Bs, tid);
  int Bs2; float aboveS2;
  descend2048(h2s, fmaxf(t - aboveS, 1e-30f), spart, bcf, bci, tid, &Bs2, &aboveS2);
  int ns = h2c[Bs2]; if (ns < 1) ns = 1;
  float vs = h2s[Bs2] / (float)ns; if (!(vs > 0.0f)) vs = 1e-30f;
  int j = (int)floorf((t - aboveS - aboveS2) / vs);  // ordinal in tie class
  if (j < 0) j = 0; if (j > ns - 1) j = ns - 1;
  if (Bs == B1 && Bs2 == B2 && j > m - 1) j = m - 1; // stay inside trim

  // ---- Phase G: find j-th element of the tie class (index order) ----------
  unsigned classkey = ((unsigned)Bs << 11) | (unsigned)Bs2;  // bits [30:8]
  if (tid == 0) { bci[1] = 0; bcf[1] = fmaxf(row[0], 1e-30f); }
  const int chunk = (VOCAB + NT - 1) / NT;       // contiguous chunks -> order
  int lo = tid * chunk; if (lo > VOCAB) lo = VOCAB;
  int hi = lo + chunk;  if (hi > VOCAB) hi = VOCAB;
  int c = 0;
  for (int i = lo; i < hi; ++i)
    c += ((__float_as_uint(row[i]) >> 8) == classkey) ? 1 : 0;
  sparti[tid] = c;
  __syncthreads();
  scan_incl_i(sparti, tid);
  int incl = sparti[tid], excl = incl - c;
  if (j >= excl && j < incl) {
    int need = j - excl;
    for (int i = lo; i < hi; ++i) {
      if ((__float_as_uint(row[i]) >> 8) == classkey) {
        if (need == 0) { bci[1] = i; bcf[1] = row[i]; break; }
        --need;
      }
    }
  }
  __syncthreads();
  int   samp_idx = bci[1];
  float samp_p   = bcf[1];

  // ---- Phase H: outputs ---------------------------------------------------
  if (tid == 0) {
    const int step = step_p[0];
    const int off  = b * BEAM + bm;
    const float prev = scores[(off * sdim + (step - 1)) * NCH + ch];
    out[off * NCH + ch] = __logf(samp_p) + prev;        // scores_out
    int* iout = (int*)out;
    iout[NCH * BSZ * BEAM + off * NCH + ch] = samp_idx; // indices_out
    if (ch == 0) iout[2 * NCH * BSZ * BEAM + off] = bm; // beams_buf
  }
}

extern "C" void kernel_launch(void* const* d_in, const int* in_sizes, int n_in,
                              void* d_out, int out_size, void* d_ws, size_t ws_size,
                              hipStream_t stream) {
  (void)n_in; (void)out_size; (void)d_ws; (void)ws_size;
  const float* lprobs = (const float*)d_in[0];
  const float* scores = (const float*)d_in[1];
  const float* uvec   = (const float*)d_in[2];
  const int*   step   = (const int*)d_in[3];
  int sdim = in_sizes[1] / (BSZ * BEAM * NCH);   // = 5 (step dim of scores)
  hipLaunchKernelGGL(topp_sample_kernel,
                     dim3(NCH * BSZ * BEAM), dim3(NT), SMEM_BYTES, stream,
                     lprobs, scores, uvec, step, (float*)d_out, sdim);
}